// Decoder_43078521979490
// MI455X (gfx1250) — compile-verified
//
#include <hip/hip_runtime.h>
#include <math.h>

typedef __attribute__((ext_vector_type(2))) float v2f;
typedef __attribute__((ext_vector_type(8))) float v8f;

#define BATCH 8192

// ---------------------------------------------------------------------------
// B-spline basis (K=3, G=5, uniform grid h=0.4, knots t[r] = (r-3)*0.4 - 1)
// out[0..7] = cubic basis values, out[8] = silu(x)
// ---------------------------------------------------------------------------
__device__ __forceinline__ void bspline9(float x, float* out) {
    const float h = 0.4f;
    float B[11];
#pragma unroll
    for (int r = 0; r < 11; ++r) {
        float t0 = (float)(r - 3) * h - 1.0f;
        float t1 = (float)(r - 2) * h - 1.0f;
        B[r] = (x >= t0 && x < t1) ? 1.0f : 0.0f;
    }
#pragma unroll
    for (int j = 1; j <= 3; ++j) {
        float inv = 1.0f / ((float)j * h);
#pragma unroll
        for (int r = 0; r < 10; ++r) {
            if (r < 11 - j) {
                float tr   = (float)(r - 3) * h - 1.0f;      // t(r)
                float trj1 = (float)(r + j - 2) * h - 1.0f;  // t(r+j+1)
                B[r] = (x - tr) * inv * B[r] + (trj1 - x) * inv * B[r + 1];
            }
        }
    }
#pragma unroll
    for (int r = 0; r < 8; ++r) out[r] = B[r];
    out[8] = x / (1.0f + __expf(-x));   // silu
}

// ---------------------------------------------------------------------------
// Pack weights: Wc[(i*9+j)*OUT + o] = coef[i][o][j]*ss[i][o] (j<8), sb (j==8)
// ---------------------------------------------------------------------------
__global__ void pack_w_kernel(const float* __restrict__ coef,
                              const float* __restrict__ sb,
                              const float* __restrict__ ss,
                              float* __restrict__ Wc, int IN, int OUT) {
    int idx = blockIdx.x * blockDim.x + threadIdx.x;
    if (idx >= IN * OUT) return;
    int i = idx / OUT, o = idx - i * OUT;
    float s = ss[i * OUT + o];
    const float* c = coef + (size_t)(i * OUT + o) * 8;
#pragma unroll
    for (int j = 0; j < 8; ++j)
        Wc[(size_t)(i * 9 + j) * OUT + o] = c[j] * s;
    Wc[(size_t)(i * 9 + 8) * OUT + o] = sb[i * OUT + o];
}

// ---------------------------------------------------------------------------
// Fused KAN layer: Y[b,o] = ns[o]*( [basis(X),silu(X)]·Wc ) + nb[o]
// Block tile 64(M) x 128(N), 8 waves, wave tile 32x32, K chunk = 4 feats * 9.
// As[m][k] row-major (stride 38), Bs[n][k] transposed (stride 38) so both A
// and B fragments are single aligned ds_load_b64s. Register double-buffering
// overlaps next chunk's global loads with the current chunk's WMMAs.
// ---------------------------------------------------------------------------
template <int IN, int OUT>
__global__ __launch_bounds__(256) void kan_gemm_kernel(
    const float* __restrict__ X, const float* __restrict__ Wc,
    const float* __restrict__ ns, const float* __restrict__ nb,
    float* __restrict__ Y) {
    constexpr int BM = 64, BN = 128, CI = 4, KC = CI * 9;  // KC = 36
    constexpr int ASTR = 38;   // even pad: 8B-aligned v2f, gcd(38,64)=2 -> no conflicts
    constexpr int BSTR = 38;
    constexpr int NW = (KC * BN) / 256;  // 18 W floats per thread
    constexpr int NC = IN / CI;
    __shared__ float As[BM * ASTR];
    __shared__ float Bs[BN * BSTR];

    const int tid  = threadIdx.x;
    const int lane = tid & 31;
    const int wave = tid >> 5;
    const int wm   = (wave >> 2) * 32;   // 0 | 32
    const int wn   = (wave & 3) * 32;    // 0,32,64,96
    const int l16  = lane & 15;
    const int lh   = lane >> 4;          // 0 | 1
    const int mblk = blockIdx.y * BM;
    const int nblk = blockIdx.x * BN;

    v8f acc00 = {}, acc01 = {}, acc10 = {}, acc11 = {};

    const int am = tid >> 2;   // 0..63 : row for A staging
    const int af = tid & 3;    // 0..3  : feature within chunk

    // ---- prefetch chunk 0 into registers ----
    float xv = X[(size_t)(mblk + am) * IN + af];
    float wreg[NW];
    {
        const float* wrow = Wc + nblk;
#pragma unroll
        for (int it = 0; it < NW; ++it) {
            int idx = it * 256 + tid;
            int r = idx >> 7, col = idx & 127;
            wreg[it] = wrow[(size_t)r * OUT + col];
        }
    }

    for (int c = 0; c < NC; ++c) {
        // ---- commit staged chunk to LDS ----
        float b9[9];
        bspline9(xv, b9);
#pragma unroll
        for (int j = 0; j < 9; ++j) As[am * ASTR + af * 9 + j] = b9[j];
#pragma unroll
        for (int it = 0; it < NW; ++it) {
            int idx = it * 256 + tid;
            int r = idx >> 7, col = idx & 127;
            Bs[col * BSTR + r] = wreg[it];
        }
        __syncthreads();

        // ---- prefetch chunk c+1 (overlaps with MMA below) ----
        if (c + 1 < NC) {
            xv = X[(size_t)(mblk + am) * IN + (c + 1) * CI + af];
            const float* wrow = Wc + (size_t)((c + 1) * KC) * OUT + nblk;
#pragma unroll
            for (int it = 0; it < NW; ++it) {
                int idx = it * 256 + tid;
                int r = idx >> 7, col = idx & 127;
                wreg[it] = wrow[(size_t)r * OUT + col];
            }
        }
        if (c + 2 < NC) {  // warm L2 two chunks ahead (global_prefetch_b8)
            __builtin_prefetch(Wc + (size_t)((c + 2) * KC) * OUT + nblk + (tid & 127), 0, 0);
        }

        // ---- MMA: 9 K-steps of 4 ----
#pragma unroll
        for (int ks = 0; ks < KC / 4; ++ks) {
            const int k0 = ks * 4 + 2 * lh;   // even -> 8B aligned
            v2f a0 = *(const v2f*)&As[(wm + l16) * ASTR + k0];
            v2f a1 = *(const v2f*)&As[(wm + 16 + l16) * ASTR + k0];
            v2f b0 = *(const v2f*)&Bs[(wn + l16) * BSTR + k0];
            v2f b1 = *(const v2f*)&Bs[(wn + 16 + l16) * BSTR + k0];
            acc00 = __builtin_amdgcn_wmma_f32_16x16x4_f32(false, a0, false, b0,
                                                          (short)0, acc00, false, false);
            acc01 = __builtin_amdgcn_wmma_f32_16x16x4_f32(false, a0, false, b1,
                                                          (short)0, acc01, false, false);
            acc10 = __builtin_amdgcn_wmma_f32_16x16x4_f32(false, a1, false, b0,
                                                          (short)0, acc10, false, false);
            acc11 = __builtin_amdgcn_wmma_f32_16x16x4_f32(false, a1, false, b1,
                                                          (short)0, acc11, false, false);
        }
        __syncthreads();
    }

    // ---- epilogue: y = ns*acc + nb ----
    const int n0 = nblk + wn + l16;
    const int n1 = n0 + 16;
    const float s0 = ns[n0], o0 = nb[n0];
    const float s1 = ns[n1], o1 = nb[n1];
#pragma unroll
    for (int r = 0; r < 8; ++r) {
        int row = mblk + wm + r + 8 * lh;          // C/D layout: lanes16-31 -> M+8
        Y[(size_t)row * OUT + n0]        = s0 * acc00[r] + o0;
        Y[(size_t)row * OUT + n1]        = s1 * acc01[r] + o1;
        Y[(size_t)(row + 16) * OUT + n0] = s0 * acc10[r] + o0;
        Y[(size_t)(row + 16) * OUT + n1] = s1 * acc11[r] + o1;
    }
}

__global__ void copy_sigma_kernel(const float* __restrict__ s, float* __restrict__ dst, int n) {
    int i = blockIdx.x * blockDim.x + threadIdx.x;
    if (i < n) dst[i] = s[i];
}

// ---------------------------------------------------------------------------
extern "C" void kernel_launch(void* const* d_in, const int* in_sizes, int n_in,
                              void* d_out, int out_size, void* d_ws, size_t ws_size,
                              hipStream_t stream) {
    const float* x     = (const float*)d_in[0];
    const float* coef0 = (const float*)d_in[2];
    const float* sb0   = (const float*)d_in[3];
    const float* ss0   = (const float*)d_in[4];
    const float* ns0   = (const float*)d_in[5];
    const float* nb0   = (const float*)d_in[6];
    const float* coef1 = (const float*)d_in[8];
    const float* sb1   = (const float*)d_in[9];
    const float* ss1   = (const float*)d_in[10];
    const float* ns1   = (const float*)d_in[11];
    const float* nb1   = (const float*)d_in[12];
    const float* coef2 = (const float*)d_in[14];
    const float* sb2   = (const float*)d_in[15];
    const float* ss2   = (const float*)d_in[16];
    const float* ns2   = (const float*)d_in[17];
    const float* nb2   = (const float*)d_in[18];
    const float* sigma = (const float*)d_in[19];

    float* ws = (float*)d_ws;
    // workspace layout (floats)
    float* h0 = ws;                               // 8192*1024
    float* h1 = h0 + (size_t)BATCH * 1024;        // 8192*512
    float* W0 = h1 + (size_t)BATCH * 512;         // 2304*1024
    float* W1 = W0 + (size_t)2304 * 1024;         // 9216*512
    float* W2 = W1 + (size_t)9216 * 512;          // 4608*512

    float* feat = (float*)d_out;                  // 8192*512
    float* sig_out = feat + (size_t)BATCH * 512;  // 512

    // pack weights (serialized on stream ahead of the GEMMs)
    pack_w_kernel<<<(256 * 1024 + 255) / 256, 256, 0, stream>>>(coef0, sb0, ss0, W0, 256, 1024);
    pack_w_kernel<<<(1024 * 512 + 255) / 256, 256, 0, stream>>>(coef1, sb1, ss1, W1, 1024, 512);
    pack_w_kernel<<<(512 * 512 + 255) / 256, 256, 0, stream>>>(coef2, sb2, ss2, W2, 512, 512);

    // layer 0: 256 -> 1024
    kan_gemm_kernel<256, 1024><<<dim3(1024 / 128, BATCH / 64), 256, 0, stream>>>(x, W0, ns0, nb0, h0);
    // layer 1: 1024 -> 512
    kan_gemm_kernel<1024, 512><<<dim3(512 / 128, BATCH / 64), 256, 0, stream>>>(h0, W1, ns1, nb1, h1);
    // layer 2: 512 -> 512 (writes feature output)
    kan_gemm_kernel<512, 512><<<dim3(512 / 128, BATCH / 64), 256, 0, stream>>>(h1, W2, ns2, nb2, feat);

    copy_sigma_kernel<<<2, 256, 0, stream>>>(sigma, sig_out, 512);
}